// NewTable_1185410973915
// MI455X (gfx1250) — compile-verified
//
#include <hip/hip_runtime.h>
#include <hip/hip_fp16.h>
#include <stdint.h>

// ---------------------------------------------------------------------------
// fp16 piecewise-linear LUT GELU, MI455X (gfx1250).
// Roofline: 268 MB traffic @ 23.3 TB/s => ~11.5 us floor => ~14 VALU/element
// budget. Eval chain: interval class from fp16 exponent bits (cut points are
// powers of 2) -> one fp32 FMA g = M*x + B (exact in interior regions: ms[]
// are powers of two, x-cp is Sterbenz-exact) -> packed {l16,d16} dword LUT ->
// fp16 mul + add (reference's double rounding preserved).
// Data movement: 4-deep GLOBAL_LOAD_ASYNC_TO_LDS_B128 pipeline (ASYNCcnt),
// 2 x 512B contiguous bursts per wave per tile, b128 stores.
// ---------------------------------------------------------------------------

#define TPB            256
#define STAGES         4
#define SUB_BYTES      (TPB * 16)            // one b128 per thread: 4 KiB
#define TILE_BYTES     (2 * SUB_BYTES)       // 8 KiB per stage (16 elems/thread)
#define ELEMS_PER_TILE (TPB * 16)            // 4096
#define SUB_ELEMS      (TPB * 8)             // 2048

typedef _Float16 f16;

__device__ __forceinline__ void async_copy_b128(uint32_t lds_off, const void* g) {
    asm volatile("global_load_async_to_lds_b128 %0, %1, off"
                 :: "v"(lds_off), "v"((uint64_t)(uintptr_t)g) : "memory");
}
// Slot reuse: drain DS reads first (async LDS writes are unordered vs DS ops).
__device__ __forceinline__ void async_copy_b128_reuse(uint32_t lds_off, const void* g) {
    asm volatile("s_wait_dscnt 0\n\t"
                 "global_load_async_to_lds_b128 %0, %1, off"
                 :: "v"(lds_off), "v"((uint64_t)(uintptr_t)g) : "memory");
}
template <int N>
__device__ __forceinline__ void wait_asynccnt() {
    asm volatile("s_wait_asynccnt %0" :: "n"(N) : "memory");
}

// hb: 16-bit fp16 pattern in low bits. cls: 20 x {M,B}. pairs: 35 x {l16,d16}.
__device__ __forceinline__ f16 eval_fast(uint32_t hb, const float2* cls,
                                         const uint32_t* pairs, f16 t0, f16 t34) {
    union { uint16_t us; f16 h; } cv; cv.us = (uint16_t)hb;
    float xf = (float)cv.h;                       // v_cvt_f32_f16
    int e  = (int)((hb >> 10) & 31u);             // v_bfe
    int ap = e < 13 ? 13 : (e > 17 ? 17 : e);     // v_med3_i32
    int j  = ((ap - 13) << 1) | (int)((hb >> 15) & 1u);
    float2 mb = cls[j];                           // ds_load_b64 (40 dwords -> conflict-free)
    float g  = fmaf(xf, mb.x, mb.y);              // lut index + decimal, fused
    float fl = floorf(g);
    int lut  = (int)fl;
    lut = lut < 0 ? 0 : (lut > 33 ? 33 : lut);    // v_med3 (guards inf garbage)
    f16 dec  = (f16)(g - fl);                     // exact in interior regions
    union { uint32_t u; f16 h[2]; } pu; pu.u = pairs[lut];  // ds_load_b32
    f16 y = pu.h[0] + pu.h[1] * dec;              // mul then add: two roundings
    if (xf <= -65504.0f) y = t0;                  // catches -inf
    if (xf >=  65504.0f) y = t34;                 // catches +inf
    if (xf != xf)        y = (f16)0.0f;           // NaN -> gelu(0) = 0
    return y;
}

__device__ __forceinline__ uint32_t eval_pair(uint32_t w, const float2* cls,
                                              const uint32_t* pairs, f16 t0, f16 t34) {
    union { uint32_t u; f16 h[2]; } po;
    po.h[0] = eval_fast(w,       cls, pairs, t0, t34);
    po.h[1] = eval_fast(w >> 16, cls, pairs, t0, t34);
    return po.u;
}

__global__ void __launch_bounds__(TPB)
gelu_lut_async_kernel(const f16* __restrict__ x,
                      const f16* __restrict__ cut_points,
                      const float* __restrict__ table,
                      const float* __restrict__ mul_scale,
                      f16* __restrict__ out,
                      int numTiles) {
    __shared__ alignas(16) uint8_t stage[STAGES * TILE_BYTES];
    __shared__ float2   s_cls[20];   // [(clamped_exp-13)*2 | sign] -> {M, B}
    __shared__ uint32_t s_pair[35];  // {lo: fp16(tab[i]), hi: fp16(tab[i+1])-fp16(tab[i])}

    const int tid = threadIdx.x;
    if (tid < 35) {
        f16 l = (f16)table[tid];
        f16 d = (f16)0.0f;
        if (tid < 34) d = (f16)table[tid + 1] - l;   // reference's fp16 interval
        union { uint32_t u; f16 h[2]; } pk; pk.h[0] = l; pk.h[1] = d;
        s_pair[tid] = pk.u;
    }
    if (tid >= 64 && tid < 84) {
        int t  = tid - 64;
        int aa = 13 + (t >> 1);                      // clamped exponent 13..17
        int sg = t & 1;
        int ci = sg ? (17 - aa) : (aa - 8);          // interval index 0..9
        float M    = mul_scale[ci];                  // exactly fp16(ms), power of 2
        float base = (ci == 0) ? 0.0f : (float)(4 * ci - 3);
        float B    = fmaf(-(float)cut_points[ci], M, base);  // exact in fp32
        s_cls[t] = make_float2(M, B);
    }
    __syncthreads();

    union { uint32_t u; f16 h[2]; } tt;
    tt.u = s_pair[0];  const f16 t0v  = tt.h[0];
    tt.u = s_pair[34]; const f16 t34v = tt.h[0];

    const int tile0   = blockIdx.x;
    const int tstride = gridDim.x;
    if (tile0 >= numTiles) return;
    const int nMine = (numTiles - tile0 + tstride - 1) / tstride;

    const size_t stride_elems = (size_t)tstride * ELEMS_PER_TILE;
    const f16* gin  = x   + ((size_t)tile0 * TPB + tid) * 8;   // sub-tile 0 lane addr
    f16*       gout = out + ((size_t)tile0 * TPB + tid) * 8;
    const uint32_t ldsbase = (uint32_t)(uintptr_t)&stage[tid * 16];

    // Fill pipeline: tile s -> slot s; two contiguous 4KB bursts per tile.
    #pragma unroll
    for (int s = 0; s < STAGES; ++s) {
        if (s < nMine) {
            const f16* p = gin + (size_t)s * stride_elems;
            async_copy_b128(ldsbase + s * TILE_BYTES,             p);
            async_copy_b128(ldsbase + s * TILE_BYTES + SUB_BYTES, p + SUB_ELEMS);
        }
    }

    const f16* gpre = gin + (size_t)STAGES * stride_elems;
    for (int i = 0; i < nMine; ++i) {
        const bool more = (i + STAGES) < nMine;
        if (more) wait_asynccnt<2 * STAGES - 2>();   // oldest tile's 2 copies done
        else      wait_asynccnt<0>();                // drain tail

        const int slot = i & (STAGES - 1);
        union { uint4 u; uint32_t w[4]; } iv0, iv1, ov0, ov1;
        iv0.u = *(const uint4*)&stage[slot * TILE_BYTES + tid * 16];
        iv1.u = *(const uint4*)&stage[slot * TILE_BYTES + SUB_BYTES + tid * 16];
        #pragma unroll
        for (int k = 0; k < 4; ++k) {
            ov0.w[k] = eval_pair(iv0.w[k], s_cls, s_pair, t0v, t34v);
            ov1.w[k] = eval_pair(iv1.w[k], s_cls, s_pair, t0v, t34v);
        }
        *(uint4*)gout               = ov0.u;
        *(uint4*)(gout + SUB_ELEMS) = ov1.u;
        gout += stride_elems;

        if (more) {
            async_copy_b128_reuse(ldsbase + slot * TILE_BYTES,             gpre);
            async_copy_b128      (ldsbase + slot * TILE_BYTES + SUB_BYTES, gpre + SUB_ELEMS);
            gpre += stride_elems;
        }
    }
}

// Scalar tail (n % 4096 elements): straightforward, reads tables from global.
__device__ __forceinline__ f16 gelu_ref_scalar(f16 h, const f16* cp,
                                               const float* tab, const float* ms) {
    if (h != h) h = (f16)0.0f;
    float xf = (float)h;
    int ci = (int)(xf >= -4.0f) + (int)(xf >= -2.0f) + (int)(xf >= -1.0f) +
             (int)(xf >= -0.5f) + (int)(xf >= 0.0f) + (int)(xf >= 0.5f) +
             (int)(xf >= 1.0f) + (int)(xf >= 2.0f) + (int)(xf >= 4.0f);
    f16 dval = h - cp[ci];
    f16 temp = dval * (f16)ms[ci];
    int idx = (int)floorf((float)temp);
    if (ci == 9 && idx == 1) idx = 0;
    f16 dec = temp - (f16)idx;
    int lut = (ci == 0) ? idx : (1 + (ci - 1) * 4 + idx);
    lut = lut < 0 ? 0 : (lut > 33 ? 33 : lut);
    f16 l = (f16)tab[lut];
    f16 r = (f16)tab[lut + 1];
    f16 y = l + (r - l) * dec;
    if (xf <= -65504.0f) y = (f16)tab[0];
    if (xf >=  65504.0f) y = (f16)tab[34];
    return y;
}

__global__ void __launch_bounds__(TPB)
gelu_lut_tail_kernel(const f16* __restrict__ x, const f16* __restrict__ cp,
                     const float* __restrict__ tab, const float* __restrict__ ms,
                     f16* __restrict__ out, long long start, long long n) {
    long long i = start + (long long)blockIdx.x * blockDim.x + threadIdx.x;
    if (i < n) out[i] = gelu_ref_scalar(x[i], cp, tab, ms);
}

extern "C" void kernel_launch(void* const* d_in, const int* in_sizes, int n_in,
                              void* d_out, int out_size, void* d_ws, size_t ws_size,
                              hipStream_t stream) {
    const f16*   x   = (const f16*)d_in[0];
    const f16*   cp  = (const f16*)d_in[1];
    const float* tab = (const float*)d_in[2];
    const float* ms  = (const float*)d_in[3];
    f16*         out = (f16*)d_out;

    const long long n        = in_sizes[0];
    const long long numTiles = n / ELEMS_PER_TILE;

    if (numTiles > 0) {
        long long blocks = numTiles < 2048 ? numTiles : 2048;
        gelu_lut_async_kernel<<<dim3((unsigned)blocks), dim3(TPB), 0, stream>>>(
            x, cp, tab, ms, out, (int)numTiles);
    }
    const long long start = numTiles * ELEMS_PER_TILE;
    const long long rem   = n - start;
    if (rem > 0) {
        long long blocks = (rem + TPB - 1) / TPB;
        gelu_lut_tail_kernel<<<dim3((unsigned)blocks), dim3(TPB), 0, stream>>>(
            x, cp, tab, ms, out, start, n);
    }
}